// DeepGRL_84808424227048
// MI455X (gfx1250) — compile-verified
//
#include <hip/hip_runtime.h>
#include <cstdint>

// ---------------------------------------------------------------------------
// GCN forward on gfx1250: fp32 WMMA GEMMs (16x16x4) with async-to-LDS A-tile
// staging, L2-resident edge scatter with hardware global_atomic_add_f32,
// fused BN+ReLU.
// ---------------------------------------------------------------------------

typedef float v2f __attribute__((ext_vector_type(2)));
typedef float v8f __attribute__((ext_vector_type(8)));

__device__ __forceinline__ void atomic_fadd(float* p, float v) {
    // Non-returning hardware fp32 atomic (near-atomic in L2). No CAS loop.
    asm volatile("global_atomic_add_f32 %0, %1, off" :: "v"(p), "v"(v) : "memory");
}

// ---------------------------------------------------------------------------
// Degree / normalization
// ---------------------------------------------------------------------------
__global__ void k_init_deg(unsigned* deg, int n) {
    int i = blockIdx.x * blockDim.x + threadIdx.x;
    if (i < n) deg[i] = 1u;                       // self loop
}

__global__ void k_count_deg(const long long* __restrict__ dst, unsigned* deg, int e) {
    int i = blockIdx.x * blockDim.x + threadIdx.x;
    if (i < e) atomicAdd(&deg[(int)dst[i]], 1u);
}

__global__ void k_dinv(const unsigned* __restrict__ deg, float* __restrict__ dinv, int n) {
    int i = blockIdx.x * blockDim.x + threadIdx.x;
    if (i < n) dinv[i] = rsqrtf((float)deg[i]);
}

// ---------------------------------------------------------------------------
// C[M,NO] = A[M,K] @ B[K,NO], fp32 WMMA 16x16x4. M multiple of 16.
// Block = NO/16 waves, covers 16 rows x NO cols. The 16xK A tile is DMA'd
// into LDS once per block (GLOBAL_LOAD_ASYNC_TO_LDS_B128, ASYNCcnt-tracked),
// then shared by all waves; B fragments stream from L2.
// LDS row stride K+4 floats -> ds_load_b64 bank = 4*m16 + k + 2*half:
// all 32 lanes land on distinct bank pairs (conflict-free).
// A-frag (16x4 f32): lanes 0-15 rows, v0/v1 = K {0,1} (half0) / {2,3} (half1).
// B-frag (4x16 f32): lanes 0-15 cols, v0 = K0|K2, v1 = K1|K3 per lane-half.
// C/D (16x16 f32):  VGPR i -> row i (lanes 0-15) / row i+8 (lanes 16-31).
// ---------------------------------------------------------------------------
template <int K, int NO>
__global__ __launch_bounds__(32 * (NO / 16)) void k_gemm_wmma(
        const float* __restrict__ A, const float* __restrict__ B,
        float* __restrict__ C, int M) {
    (void)M;
    constexpr int NT   = NO / 16;          // waves per block
    constexpr int LSTR = K + 4;            // padded LDS row stride (floats)
    __shared__ __align__(16) float tileA[16 * LSTR];

    const int tid  = threadIdx.x;
    const int wave = tid >> 5;
    const int lane = tid & 31;
    const int half = lane >> 4;
    const int m16  = lane & 15;
    const int r0   = blockIdx.x * 16;
    const int c0   = wave * 16;

    // ---- async-stage 16xK fp32 A tile into LDS (16B chunks) ----
    constexpr int CPR    = K / 4;          // chunks per row
    constexpr int CHUNKS = 16 * CPR;
    for (int c = tid; c < CHUNKS; c += 32 * NT) {
        const int row  = c / CPR;
        const int colf = (c % CPR) * 4;
        const float* gsrc = A + (size_t)(r0 + row) * K + colf;
        unsigned ldsoff = (unsigned)(size_t)&tileA[row * LSTR + colf];
        asm volatile("global_load_async_to_lds_b128 %0, %1, off"
                     :: "v"(ldsoff), "v"(gsrc) : "memory");
    }
    asm volatile("s_wait_asynccnt 0x0" ::: "memory");
    __syncthreads();

    // ---- K-loop: WMMA accumulate ----
    const float* arow = &tileA[m16 * LSTR + 2 * half];
    const float* bcol = B + (size_t)(2 * half) * NO + (c0 + m16);
    v8f acc = {};
#pragma unroll
    for (int k = 0; k < K; k += 4) {
        v2f a = *reinterpret_cast<const v2f*>(arow + k);   // ds_load_b64
        v2f b;
        b.x = bcol[(size_t)k * NO];
        b.y = bcol[(size_t)(k + 1) * NO];
        acc = __builtin_amdgcn_wmma_f32_16x16x4_f32(
            /*neg_a=*/false, a, /*neg_b=*/false, b,
            /*c_mod=*/(short)0, acc, /*reuse_a=*/false, /*reuse_b=*/false);
    }

    float* crow = C + (size_t)r0 * NO + c0 + m16;
#pragma unroll
    for (int i = 0; i < 8; ++i)
        crow[(size_t)(i + 8 * half) * NO] = acc[i];
}

// ---------------------------------------------------------------------------
// O[i,:] = dinv[i]^2 * H[i,:] + bias   (self-loop term + bias of GCNConv)
// ---------------------------------------------------------------------------
template <int F>
__global__ void k_self_init(const float* __restrict__ H, const float* __restrict__ dinv,
                            const float* __restrict__ bias, float* __restrict__ O, int n) {
    int idx = blockIdx.x * blockDim.x + threadIdx.x;
    if (idx >= n * F) return;
    int i = idx / F;
    int f = idx % F;
    float di = dinv[i];
    O[idx] = di * di * H[idx] + bias[f];
}

// ---------------------------------------------------------------------------
// Edge scatter: one wave per edge; lane carries F/32 contiguous features.
// O[dst,:] += dinv[src]*dinv[dst] * H[src,:]   via hardware fp32 atomics (L2).
// ---------------------------------------------------------------------------
template <int F>
__global__ void k_edge_agg(const long long* __restrict__ src,
                           const long long* __restrict__ dst,
                           const float* __restrict__ dinv,
                           const float* __restrict__ H,
                           float* __restrict__ O, int e) {
    constexpr int V = F / 32;
    int wid  = blockIdx.x * (blockDim.x >> 5) + (threadIdx.x >> 5);
    int lane = threadIdx.x & 31;
    if (wid >= e) return;
    int s = (int)src[wid];
    int d = (int)dst[wid];
    float w = dinv[s] * dinv[d];
    const float* hs = H + (size_t)s * F + lane * V;
    float*       od = O + (size_t)d * F + lane * V;
    float hv[V];
#pragma unroll
    for (int j = 0; j < V; ++j) hv[j] = hs[j];           // merges to b128/b64 load
#pragma unroll
    for (int j = 0; j < V; ++j) atomic_fadd(od + j, w * hv[j]);
}

// ---------------------------------------------------------------------------
// BatchNorm (training-mode, biased variance) over nodes, F=128.
// ---------------------------------------------------------------------------
__global__ void k_zero_stats(float* stats) { stats[threadIdx.x] = 0.0f; }

__global__ void k_bn_stats(const float* __restrict__ X, float* __restrict__ stats, int n) {
    const int f  = threadIdx.x;               // 0..127
    const int r0 = blockIdx.x * 256;
    const int r1 = (r0 + 256 < n) ? (r0 + 256) : n;
    float s = 0.0f, ss = 0.0f;
    for (int r = r0; r < r1; ++r) {
        float x = X[(size_t)r * 128 + f];     // coalesced across lanes
        s += x;
        ss += x * x;
    }
    atomic_fadd(&stats[f], s);
    atomic_fadd(&stats[128 + f], ss);
}

__global__ void k_bn_apply(float* __restrict__ X, const float* __restrict__ stats,
                           const float* __restrict__ g, const float* __restrict__ be, int n) {
    int idx = blockIdx.x * blockDim.x + threadIdx.x;
    if (idx >= n * 128) return;
    int f = idx & 127;
    float invn = 1.0f / (float)n;
    float mean = stats[f] * invn;
    float var  = stats[128 + f] * invn - mean * mean;
    float inv  = rsqrtf(var + 1e-5f);
    float y    = g[f] * (X[idx] - mean) * inv + be[f];
    X[idx] = fmaxf(y, 0.0f);
}

// ---------------------------------------------------------------------------
// Host orchestration (all on `stream`; graph-capture safe).
// ---------------------------------------------------------------------------
extern "C" void kernel_launch(void* const* d_in, const int* in_sizes, int n_in,
                              void* d_out, int out_size, void* d_ws, size_t ws_size,
                              hipStream_t stream) {
    (void)n_in; (void)out_size; (void)ws_size;
    const float*     x   = (const float*)d_in[0];
    const long long* ei  = (const long long*)d_in[1];   // int64 edge_index [2,E]
    const float*     W1  = (const float*)d_in[2];
    const float*     b1  = (const float*)d_in[3];
    const float*     g1  = (const float*)d_in[4];
    const float*     be1 = (const float*)d_in[5];
    const float*     W2  = (const float*)d_in[6];
    const float*     b2  = (const float*)d_in[7];
    const float*     g2  = (const float*)d_in[8];
    const float*     be2 = (const float*)d_in[9];
    const float*     W3  = (const float*)d_in[10];
    const float*     b3  = (const float*)d_in[11];
    float*           out = (float*)d_out;

    const int N = in_sizes[0] / 128;   // 50000 (multiple of 16)
    const int E = in_sizes[1] / 2;     // 600000
    const long long* src = ei;
    const long long* dst = ei + E;

    // Workspace carve-out
    auto al = [](size_t o) { return (o + 255) & ~(size_t)255; };
    char* ws = (char*)d_ws;
    size_t off = 0;
    float*    dinv  = (float*)(ws + off);    off = al(off + (size_t)N * 4);
    unsigned* deg   = (unsigned*)(ws + off); off = al(off + (size_t)N * 4);
    float*    stats = (float*)(ws + off);    off = al(off + 256 * 4);
    float*    bufA  = (float*)(ws + off);    off = al(off + (size_t)N * 128 * 4);
    float*    bufB  = (float*)(ws + off);    off = al(off + (size_t)N * 128 * 4);

    const int TB = 256;
    const int gN   = (N + TB - 1) / TB;
    const int gE   = (E + TB - 1) / TB;
    const int gNF  = (N * 128 + TB - 1) / TB;
    const int gNF3 = (N * 64 + TB - 1) / TB;
    const int gEw  = (E + 7) / 8;            // 8 waves per 256-thread block
    const int gBN  = (N + 255) / 256;

    // ---- normalization coefficients
    k_init_deg<<<gN, TB, 0, stream>>>(deg, N);
    k_count_deg<<<gE, TB, 0, stream>>>(dst, deg, E);
    k_dinv<<<gN, TB, 0, stream>>>(deg, dinv, N);

    // ---- layer 1: GCN(x, W1) -> BN -> ReLU
    k_gemm_wmma<128, 128><<<N / 16, 256, 0, stream>>>(x, W1, bufA, N);
    k_self_init<128><<<gNF, TB, 0, stream>>>(bufA, dinv, b1, bufB, N);
    k_edge_agg<128><<<gEw, TB, 0, stream>>>(src, dst, dinv, bufA, bufB, E);
    k_zero_stats<<<1, 256, 0, stream>>>(stats);
    k_bn_stats<<<gBN, 128, 0, stream>>>(bufB, stats, N);
    k_bn_apply<<<gNF, TB, 0, stream>>>(bufB, stats, g1, be1, N);

    // ---- layer 2: GCN(h, W2) -> BN -> ReLU
    k_gemm_wmma<128, 128><<<N / 16, 256, 0, stream>>>(bufB, W2, bufA, N);
    k_self_init<128><<<gNF, TB, 0, stream>>>(bufA, dinv, b2, bufB, N);
    k_edge_agg<128><<<gEw, TB, 0, stream>>>(src, dst, dinv, bufA, bufB, E);
    k_zero_stats<<<1, 256, 0, stream>>>(stats);
    k_bn_stats<<<gBN, 128, 0, stream>>>(bufB, stats, N);
    k_bn_apply<<<gNF, TB, 0, stream>>>(bufB, stats, g2, be2, N);

    // ---- layer 3: GCN(h, W3) -> d_out [N,64]
    k_gemm_wmma<128, 64><<<N / 16, 128, 0, stream>>>(bufB, W3, bufA, N);
    k_self_init<64><<<gNF3, TB, 0, stream>>>(bufA, dinv, b3, out, N);
    k_edge_agg<64><<<gEw, TB, 0, stream>>>(src, dst, dinv, bufA, out, E);
}